// GRAPE_50233937494018
// MI455X (gfx1250) — compile-verified
//
#include <hip/hip_runtime.h>

typedef __attribute__((ext_vector_type(2))) float v2f;
typedef __attribute__((ext_vector_type(8))) float v8f;

#define N_NODES 4096
#define NGENE 3
#define NROLE 2

// ---------------------------------------------------------------- utilities
__global__ void k_zero(float* __restrict__ p, int n) {
  int i = blockIdx.x * blockDim.x + threadIdx.x;
  if (i < n) p[i] = 0.0f;
}

// ---------------------------------------------------- s[g,r] = x @ W[g,r]
// tiny GEMM (<=0.4 GFLOP total) -> plain VALU is fine, L2 feeds it.
__global__ void k_featmm(const float* __restrict__ x,   // [4096, FIN]
                         const float* __restrict__ W,   // [NGENE*NROLE, FIN, DOUT]
                         float* __restrict__ s,         // [NGENE*NROLE, 4096, DOUT]
                         int FIN, int DOUT) {
  int tid = blockIdx.x * blockDim.x + threadIdx.x;
  int total = NGENE * NROLE * N_NODES * DOUT;
  if (tid >= total) return;
  int o  = tid % DOUT;
  int n  = (tid / DOUT) % N_NODES;
  int gr = tid / (DOUT * N_NODES);
  const float* xr = x + (size_t)n * FIN;
  const float* Wc = W + (size_t)gr * FIN * DOUT + o;
  float acc = 0.0f;
  for (int f = 0; f < FIN; ++f) acc += xr[f] * Wc[(size_t)f * DOUT];
  s[tid] = acc;
}

// --------------------------------------------------------- big aggregation
// h[g] = relu( sum_r adj[g,r] @ s[g,r] + b[g] ), plus partial sums for the
// per-gene mean (attention input). One wave = 16 rows x DOUT cols, WMMA
// f32 16x16x4. A straight from global (b64/lane), B staged in LDS
// pair-interleaved so each fragment is a single ds b64.
template <int DOUT>
__global__ void __launch_bounds__(128)
k_aggregate(const float* __restrict__ adj,   // [NGENE,NROLE,4096,4096]
            const float* __restrict__ s,     // [NGENE,NROLE,4096,DOUT]
            const float* __restrict__ bias,  // [NGENE,DOUT]
            float* __restrict__ h,           // [NGENE,4096,DOUT]
            float* __restrict__ sqacc) {     // [NGENE] mean accumulators
  constexpr int KC = 64;            // K chunk staged in LDS
  constexpr int NT = DOUT / 16;     // 16-wide N tiles per wave
  __shared__ float lds[KC * DOUT];  // ((k/2)*DOUT + n)*2 + (k&1)

  const int tid  = threadIdx.x;     // 0..127, 4 waves
  const int wave = tid >> 5;
  const int lane = tid & 31;
  const int l15  = lane & 15;
  const int hf   = lane >> 4;       // K half selector
  const int gene = blockIdx.y;
  const int row0 = blockIdx.x * 64 + wave * 16;

  v8f acc[NT] = {};

  for (int r = 0; r < NROLE; ++r) {
    const float* Abase =
        adj + ((size_t)(gene * NROLE + r) * N_NODES + row0 + l15) * N_NODES + 2 * hf;
    const float* srow = s + (size_t)(gene * NROLE + r) * N_NODES * DOUT;

    for (int k0 = 0; k0 < N_NODES; k0 += KC) {
      __syncthreads();
      // cooperative fill: coalesced global read, pair-interleaved LDS write
      for (int e = tid; e < KC * DOUT; e += 128) {
        int kk = e / DOUT;
        int n  = e - kk * DOUT;
        lds[((kk >> 1) * DOUT + n) * 2 + (kk & 1)] = srow[(size_t)k0 * DOUT + e];
      }
      __syncthreads();

      __builtin_prefetch(Abase + k0 + KC, 0, 1);  // next adj chunk -> global_prefetch_b8

#pragma unroll
      for (int kk = 0; kk < KC; kk += 4) {
        v2f a = *(const v2f*)(Abase + k0 + kk);   // A frag: 2 VGPRs / lane
#pragma unroll
        for (int t = 0; t < NT; ++t) {
          v2f bf = *(const v2f*)&lds[(((kk >> 1) + hf) * DOUT + t * 16 + l15) * 2];
          acc[t] = __builtin_amdgcn_wmma_f32_16x16x4_f32(
              false, a, false, bf, (short)0, acc[t], false, false);
        }
      }
    }
  }

  // epilogue: bias + relu + store + partial sum for mean
  const float* bg = bias + gene * DOUT;
  float lsum = 0.0f;
#pragma unroll
  for (int t = 0; t < NT; ++t) {
#pragma unroll
    for (int v = 0; v < 8; ++v) {
      int row = row0 + v + 8 * hf;      // C/D layout: lanes16-31 -> M = v+8
      int col = t * 16 + l15;
      float val = fmaxf(acc[t][v] + bg[col], 0.0f);
      h[((size_t)gene * N_NODES + row) * DOUT + col] = val;
      lsum += val;
    }
  }
  atomicAdd(&sqacc[gene], lsum);
}

// ------------------------------------------------- 3x3 attention + softmax
__global__ void k_attention(const float* __restrict__ sqacc,  // [NGENE] sums
                            const float* __restrict__ Wk,     // [3,3] (this layer)
                            const float* __restrict__ Wq,     // [3,3]
                            float* __restrict__ a_out,        // [NGENE]
                            float inv_count) {
  if (threadIdx.x != 0 || blockIdx.x != 0) return;
  float sq[NGENE], kk[NGENE], e[NGENE];
  for (int g = 0; g < NGENE; ++g) sq[g] = sqacc[g] * inv_count;
  for (int j = 0; j < NGENE; ++j) {
    float a = 0.0f;
    for (int g = 0; g < NGENE; ++g) a += sq[g] * Wk[g * NGENE + j];
    kk[j] = fmaxf(a, 0.0f);
  }
  float m = -1e30f;
  for (int j = 0; j < NGENE; ++j) {
    float a = 0.0f;
    for (int i = 0; i < NGENE; ++i) a += kk[i] * Wq[i * NGENE + j];
    e[j] = a;
    m = fmaxf(m, a);
  }
  float ssum = 0.0f;
  for (int j = 0; j < NGENE; ++j) { e[j] = __expf(e[j] - m); ssum += e[j]; }
  for (int j = 0; j < NGENE; ++j) a_out[j] = e[j] / ssum;
}

// ------------------------------------- out[n,d] = sum_g a[g] * h[g,n,d]
__global__ void k_combine(const float* __restrict__ h, const float* __restrict__ a,
                          float* __restrict__ out, int per_gene) {
  int tid = blockIdx.x * blockDim.x + threadIdx.x;
  if (tid >= per_gene) return;
  out[tid] = a[0] * h[tid] + a[1] * h[tid + (size_t)per_gene] +
             a[2] * h[tid + 2 * (size_t)per_gene];
}

// ---------------------------------------------------------------- launcher
extern "C" void kernel_launch(void* const* d_in, const int* in_sizes, int n_in,
                              void* d_out, int out_size, void* d_ws, size_t ws_size,
                              hipStream_t stream) {
  const float* x   = (const float*)d_in[0];  // [4096,64]
  const float* adj = (const float*)d_in[1];  // [3,2,4096,4096]
  const float* W1  = (const float*)d_in[2];  // [3,2,64,64]
  const float* b1  = (const float*)d_in[3];  // [3,64]
  const float* W2  = (const float*)d_in[4];  // [3,2,64,32]
  const float* b2  = (const float*)d_in[5];  // [3,32]
  const float* Wk  = (const float*)d_in[6];  // [2,3,3]
  const float* Wq  = (const float*)d_in[7];  // [2,3,3]
  float* out = (float*)d_out;                // [4096,32]

  float* ws  = (float*)d_ws;
  float* sq1 = ws;        // 4
  float* sq2 = ws + 4;    // 4
  float* a1  = ws + 8;    // 4
  float* a2  = ws + 12;   // 4
  float* s1  = ws + 16;                                   // 3*2*4096*64
  float* h1  = s1 + (size_t)NGENE * NROLE * N_NODES * 64; // 3*4096*64
  float* hc  = h1 + (size_t)NGENE * N_NODES * 64;         // 4096*64
  float* s2  = hc + (size_t)N_NODES * 64;                 // 3*2*4096*32
  float* h2  = s2 + (size_t)NGENE * NROLE * N_NODES * 32; // 3*4096*32

  k_zero<<<1, 16, 0, stream>>>(ws, 16);

  // ---------------- layer 1 (DOUT = 64) ----------------
  {
    int total = NGENE * NROLE * N_NODES * 64;
    k_featmm<<<(total + 255) / 256, 256, 0, stream>>>(x, W1, s1, 64, 64);
  }
  k_aggregate<64><<<dim3(N_NODES / 64, NGENE), 128, 0, stream>>>(adj, s1, b1, h1, sq1);
  k_attention<<<1, 32, 0, stream>>>(sq1, Wk + 0, Wq + 0, a1,
                                    1.0f / (float)(N_NODES * 64));
  {
    int total = N_NODES * 64;
    k_combine<<<(total + 255) / 256, 256, 0, stream>>>(h1, a1, hc, total);
  }

  // ---------------- layer 2 (DOUT = 32) ----------------
  {
    int total = NGENE * NROLE * N_NODES * 32;
    k_featmm<<<(total + 255) / 256, 256, 0, stream>>>(hc, W2, s2, 64, 32);
  }
  k_aggregate<32><<<dim3(N_NODES / 64, NGENE), 128, 0, stream>>>(adj, s2, b2, h2, sq2);
  k_attention<<<1, 32, 0, stream>>>(sq2, Wk + 9, Wq + 9, a2,
                                    1.0f / (float)(N_NODES * 32));
  {
    int total = N_NODES * 32;
    k_combine<<<(total + 255) / 256, 256, 0, stream>>>(h2, a2, out, total);
  }
}